// ODENet_33182917329393
// MI455X (gfx1250) — compile-verified
//
#include <hip/hip_runtime.h>

typedef float v2f __attribute__((ext_vector_type(2)));
typedef float v8f __attribute__((ext_vector_type(8)));

// Swap the two 16-lane halves of a wave32 (VALU path; fallback to shfl).
__device__ __forceinline__ float swap16(float x) {
#if __has_builtin(__builtin_amdgcn_permlanex16)
  int xi = __builtin_bit_cast(int, x);
  int r = __builtin_amdgcn_permlanex16(xi, xi, 0x76543210, 0xFEDCBA98, false, false);
  return __builtin_bit_cast(float, r);
#else
  return __shfl_xor(x, 16, 32);
#endif
}

// D-layout (VGPR i: row i+8*hi, col lane%16) -> B-layout
// (B[j], j=2c+v: row 4c+2*hi+v, col lane%16) for a 16x16 f32 matrix.
__device__ __forceinline__ v8f relayoutDtoB(v8f y, bool lo) {
  float s0 = swap16(y[0]), s1 = swap16(y[1]);
  float s2 = swap16(y[2]), s3 = swap16(y[3]);
  float s4 = swap16(y[4]), s5 = swap16(y[5]);
  float s6 = swap16(y[6]), s7 = swap16(y[7]);
  v8f b;
  b[0] = lo ? y[0] : s2;   // rows 0 / 2
  b[1] = lo ? y[1] : s3;   // rows 1 / 3
  b[2] = lo ? y[4] : s6;   // rows 4 / 6
  b[3] = lo ? y[5] : s7;   // rows 5 / 7
  b[4] = lo ? s0   : y[2]; // rows 8 / 10
  b[5] = lo ? s1   : y[3]; // rows 9 / 11
  b[6] = lo ? s4   : y[6]; // rows 12 / 14
  b[7] = lo ? s5   : y[7]; // rows 13 / 15
  return b;
}

// acc += At(16x16, A-layout chunks) * B(16x16, B-layout) via 4 chained K=4 WMMAs.
__device__ __forceinline__ v8f mm16(const float* At, v8f B, v8f acc) {
#pragma unroll
  for (int c = 0; c < 4; ++c) {
    v2f a; a[0] = At[2 * c]; a[1] = At[2 * c + 1];
    v2f b; b[0] = B[2 * c];  b[1] = B[2 * c + 1];
    acc = __builtin_amdgcn_wmma_f32_16x16x4_f32(false, a, false, b, (short)0,
                                                acc, false, false);
  }
  return acc;
}

// vf(h) = relu(h @ W1 + b1) @ W2 + b2, state in D layout (h^T).
__device__ __forceinline__ v8f vf(v8f y, const float* W1t, const float* W2t,
                                  v8f b1v, v8f b2v, bool lo) {
  v8f B = relayoutDtoB(y, lo);
  v8f t = mm16(W1t, B, b1v);
#pragma unroll
  for (int i = 0; i < 8; ++i) t[i] = fmaxf(t[i], 0.0f);
  v8f B2 = relayoutDtoB(t, lo);
  return mm16(W2t, B2, b2v);
}

__global__ void __launch_bounds__(256) odenet_dopri5_wmma(
    const float* __restrict__ x, const float* __restrict__ Win,
    const float* __restrict__ bin, const float* __restrict__ W1,
    const float* __restrict__ b1, const float* __restrict__ W2,
    const float* __restrict__ b2, const float* __restrict__ Wout,
    const float* __restrict__ bout, float* __restrict__ out) {
  const int tid  = blockIdx.x * blockDim.x + threadIdx.x;
  const int wave = tid >> 5;
  const int lane = threadIdx.x & 31;
  const int hi   = (lane >> 4) & 1;
  const int m    = lane & 15;
  const bool lo  = (hi == 0);
  const int base = wave * 16;  // 16 batch rows per wave

  // ---- Batched preload: x tile (16 x b64) + W_in^T chunks (32 x b32) +
  //      persistent ODE weights. Issue everything, then a single wait,
  //      so the cold-HBM latency is paid once, pipelined.
  float xa[32], wa[32];
  const float* xrow = x + (size_t)(base + m) * 64;
#pragma unroll
  for (int c = 0; c < 16; ++c) {
    int k0 = 4 * c + 2 * hi;
    v2f xb = *(const v2f*)(xrow + k0);  // k0 even -> 8B aligned
    xa[2 * c]     = xb[0];
    xa[2 * c + 1] = xb[1];
  }
#pragma unroll
  for (int c = 0; c < 16; ++c) {
    int k0 = 4 * c + 2 * hi;
    wa[2 * c]     = Win[k0 * 16 + m];
    wa[2 * c + 1] = Win[(k0 + 1) * 16 + m];
  }
  float W1t[8], W2t[8];
#pragma unroll
  for (int j = 0; j < 8; ++j) {
    int c = j >> 1, v = j & 1;
    int k = 4 * c + 2 * hi + v;
    W1t[j] = W1[k * 16 + m];
    W2t[j] = W2[k * 16 + m];
  }
  v8f b1v, b2v;
#pragma unroll
  for (int i = 0; i < 8; ++i) {
    int r = i + 8 * hi;
    b1v[i] = b1[r];
    b2v[i] = b2[r];
  }

  // ---- Input layer: h0^T = W_in^T @ x^T + b_in  (16 chained K=4 WMMAs).
  v8f y;
#pragma unroll
  for (int i = 0; i < 8; ++i) y[i] = bin[i + 8 * hi];
#pragma unroll
  for (int c = 0; c < 16; ++c) {
    v2f a; a[0] = wa[2 * c]; a[1] = wa[2 * c + 1];
    v2f b; b[0] = xa[2 * c]; b[1] = xa[2 * c + 1];
    y = __builtin_amdgcn_wmma_f32_16x16x4_f32(false, a, false, b, (short)0, y,
                                              false, false);
  }

  // ---- 16 fixed dopri5 steps, fully in registers.
  const float dt = 1.0f / 16.0f;
#pragma unroll 1
  for (int s = 0; s < 16; ++s) {
    v8f k1 = vf(y, W1t, W2t, b1v, b2v, lo);
    v8f t  = y + k1 * (dt * 0.2f);
    v8f k2 = vf(t, W1t, W2t, b1v, b2v, lo);
    t = y + k1 * (dt * 3.0f / 40.0f) + k2 * (dt * 9.0f / 40.0f);
    v8f k3 = vf(t, W1t, W2t, b1v, b2v, lo);
    t = y + k1 * (dt * 44.0f / 45.0f) - k2 * (dt * 56.0f / 15.0f) +
        k3 * (dt * 32.0f / 9.0f);
    v8f k4 = vf(t, W1t, W2t, b1v, b2v, lo);
    t = y + k1 * (dt * 19372.0f / 6561.0f) - k2 * (dt * 25360.0f / 2187.0f) +
        k3 * (dt * 64448.0f / 6561.0f) - k4 * (dt * 212.0f / 729.0f);
    v8f k5 = vf(t, W1t, W2t, b1v, b2v, lo);
    t = y + k1 * (dt * 9017.0f / 3168.0f) - k2 * (dt * 355.0f / 33.0f) +
        k3 * (dt * 46732.0f / 5247.0f) + k4 * (dt * 49.0f / 176.0f) -
        k5 * (dt * 5103.0f / 18656.0f);
    v8f k6 = vf(t, W1t, W2t, b1v, b2v, lo);
    y = y + k1 * (dt * 35.0f / 384.0f) + k3 * (dt * 500.0f / 1113.0f) +
        k4 * (dt * 125.0f / 192.0f) - k5 * (dt * 2187.0f / 6784.0f) +
        k6 * (dt * 11.0f / 84.0f);
  }

  // ---- Output layer (weights loaded only now, so they aren't live across
  //      the 16-step loop): out^T = W_out^T @ h^T + b_out (rows 0..1 valid).
  float Wot[8];
#pragma unroll
  for (int j = 0; j < 8; ++j) {
    int c = j >> 1, v = j & 1;
    int k = 4 * c + 2 * hi + v;
    Wot[j] = (m < 2) ? Wout[k * 2 + m] : 0.0f;  // W_out^T zero-padded
  }
  v8f obias;
#pragma unroll
  for (int i = 0; i < 8; ++i) {
    int r = i + 8 * hi;
    obias[i] = (r < 2) ? bout[r] : 0.0f;
  }
  v8f Bf = relayoutDtoB(y, lo);
  v8f o  = mm16(Wot, Bf, obias);
  if (lo) {
    v2f r; r[0] = o[0]; r[1] = o[1];
    *(v2f*)(out + (size_t)(base + m) * 2) = r;
  }
}

extern "C" void kernel_launch(void* const* d_in, const int* in_sizes, int n_in,
                              void* d_out, int out_size, void* d_ws,
                              size_t ws_size, hipStream_t stream) {
  const float* x    = (const float*)d_in[0];
  const float* Win  = (const float*)d_in[1];
  const float* bin  = (const float*)d_in[2];
  const float* W1   = (const float*)d_in[3];
  const float* b1   = (const float*)d_in[4];
  const float* W2   = (const float*)d_in[5];
  const float* b2   = (const float*)d_in[6];
  const float* Wout = (const float*)d_in[7];
  const float* bout = (const float*)d_in[8];
  float* out = (float*)d_out;

  const int B = in_sizes[0] / 64;       // 65536 rows
  const int waves = B / 16;             // one 16-row tile per wave32
  const int threads = 256;              // 8 waves per block
  const int blocks = (waves * 32) / threads;
  odenet_dopri5_wmma<<<blocks, threads, 0, stream>>>(x, Win, bin, W1, b1, W2,
                                                     b2, Wout, bout, out);
}